// NaturalCubicSpline_83966610637198
// MI455X (gfx1250) — compile-verified
//
#include <hip/hip_runtime.h>
#include <hip/hip_bf16.h>

// ---------------------------------------------------------------------------
// NaturalCubicSpline on MI455X (gfx1250)
//
// Store-bandwidth-bound: 256 MB of f32 output per call (~11 us at 23.3 TB/s).
// Coefficient tables (524 KB) are L2-resident; gathers are organized as fully
// coalesced 128 B rows (8 lanes x float4 per query, wave32).
// CDNA5-specific paths used:
//   * TENSOR_LOAD_TO_LDS (TDM DMA) stages each block's tile of `t` into LDS,
//     synchronized with s_wait_tensorcnt (6-arg clang-23 builtin form).
//   * Non-temporal 128-bit stores for the write-once output (keeps the
//     coefficient tables hot in L2/WGP$).
// WMMA is intentionally NOT used: the B operand of the would-be 1x4 * 4x32
// product is a data-dependent gather per query, so no 16-row tile can share
// a B matrix.
// ---------------------------------------------------------------------------

typedef unsigned int u32x4 __attribute__((ext_vector_type(4)));
typedef int          i32x8 __attribute__((ext_vector_type(8)));
typedef int          i32x4 __attribute__((ext_vector_type(4)));
typedef float        f32x4 __attribute__((ext_vector_type(4)));

#define SPL_BLOCK  256   // 8 waves
#define SPL_TILE_Q 256   // queries per block (1 KB of t in LDS)
#define SPL_CH     32    // channels (float4 chunks: 8 per query)

#if defined(__has_builtin)
#if __has_builtin(__builtin_amdgcn_tensor_load_to_lds) && __has_builtin(__builtin_amdgcn_s_wait_tensorcnt)
#define SPL_HAVE_TDM 1
#endif
#endif
#ifndef SPL_HAVE_TDM
#define SPL_HAVE_TDM 0
#endif

__global__ __launch_bounds__(SPL_BLOCK)
void NaturalCubicSpline_83966610637198_kernel(
    const float* __restrict__ t, const float* __restrict__ knots,
    const float* __restrict__ a, const float* __restrict__ b,
    const float* __restrict__ c, const float* __restrict__ d,
    float* __restrict__ out, int n_q, int max_idx)
{
    __shared__ float s_t[SPL_TILE_Q];

    const int base = blockIdx.x * SPL_TILE_Q;

#if SPL_HAVE_TDM
    // ---- Tensor Data Mover: DMA this block's slice of t (<=1 KB) into LDS.
    // 1D descriptor: tile_dim0 = SPL_TILE_Q, tensor_dim0 = valid remainder
    // (TDM zero-fills OOB reads, and those queries are masked in the loop).
    if (threadIdx.x < 32) {   // a single wave issues the DMA
        unsigned long long ga = (unsigned long long)(const void*)(t + base);
        unsigned int lds_off  = (unsigned int)(unsigned long long)(const void*)s_t;
        unsigned int remain   = (unsigned int)(n_q - base);
        if (remain > SPL_TILE_Q) remain = SPL_TILE_Q;

        // D# group 0: count=1 | lds_addr | global_addr[56:0] | type=2
        u32x4 g0;
        g0[0] = 1u;                                   // count=1, user descriptor
        g0[1] = lds_off;                              // lds_addr (bytes)
        g0[2] = (unsigned int)ga;                     // global_addr[31:0]
        g0[3] = (unsigned int)((ga >> 32) & 0x1FFFFFFull) | 0x80000000u; // [56:32] | type=2

        // D# group 1: data_size=4B, tensor_dim0=remain, tensor_dim1=1,
        // tile_dim0=SPL_TILE_Q, tile_dim1=1, strides trivial.
        i32x8 g1;
        g1[0] = (int)(2u << 16);                                // data_size=2 (4 bytes)
        g1[1] = (int)((remain & 0xFFFFu) << 16);                // tensor_dim0[15:0]
        g1[2] = (int)(((remain >> 16) & 0xFFFFu) | (1u << 16)); // tensor_dim0[31:16] | tensor_dim1=1
        g1[3] = (int)((unsigned)SPL_TILE_Q << 16);              // tile_dim0
        g1[4] = 1;                                              // tile_dim1=1, tile_dim2=0
        g1[5] = SPL_TILE_Q;                                     // tensor_dim0_stride (unused for 1D)
        g1[6] = 0;
        g1[7] = 0;

        i32x4 gz4 = {0, 0, 0, 0};
        i32x8 gz8 = {0, 0, 0, 0, 0, 0, 0, 0};
        // clang-23 / therock-10.0 form: 6 args (g0, g1, g2, g3, extra, cpol)
        __builtin_amdgcn_tensor_load_to_lds(g0, g1, gz4, gz4, gz8, 0);
        __builtin_amdgcn_s_wait_tensorcnt(0);
    }
#else
    {   // fallback: direct LDS fill
        int gt = base + (int)threadIdx.x;
        s_t[threadIdx.x] = (gt < n_q) ? t[gt] : 0.0f;
    }
#endif
    __syncthreads();

    // wave32 layout: 8 lanes per query, one float4 of channels per lane.
    const int qloc0 = (int)threadIdx.x >> 3;   // 0..31: query within a pass
    const int ch4   = (int)threadIdx.x & 7;    // float4 chunk 0..7

#pragma unroll
    for (int pass = 0; pass < SPL_TILE_Q / 32; ++pass) {
        const int q  = qloc0 + pass * 32;
        const int gq = base + q;
        if (gq < n_q) {
            const float tq = s_t[q];

            // searchsorted(arange(K), t, 'left') - 1, clamped — exact for
            // uniform integer knots: idx = clamp(ceil(t)-1, 0, K-2).
            int idx = (int)ceilf(tq) - 1;
            idx = idx < 0 ? 0 : idx;
            idx = idx > max_idx ? max_idx : idx;
            const float frac = tq - knots[idx];

            const size_t row = (size_t)idx * SPL_CH;
            const f32x4 av = reinterpret_cast<const f32x4*>(a + row)[ch4];
            const f32x4 bv = reinterpret_cast<const f32x4*>(b + row)[ch4];
            const f32x4 cv = reinterpret_cast<const f32x4*>(c + row)[ch4];
            const f32x4 dv = reinterpret_cast<const f32x4*>(d + row)[ch4];

            f32x4 r;
            r.x = fmaf(frac, fmaf(frac, fmaf(frac, dv.x, cv.x), bv.x), av.x);
            r.y = fmaf(frac, fmaf(frac, fmaf(frac, dv.y, cv.y), bv.y), av.y);
            r.z = fmaf(frac, fmaf(frac, fmaf(frac, dv.z, cv.z), bv.z), av.z);
            r.w = fmaf(frac, fmaf(frac, fmaf(frac, dv.w, cv.w), bv.w), av.w);

            // Write-once output: non-temporal 128-bit store (don't rinse L2).
            f32x4* dst = reinterpret_cast<f32x4*>(out + (size_t)gq * SPL_CH) + ch4;
            __builtin_nontemporal_store(r, dst);
        }
    }
}

extern "C" void kernel_launch(void* const* d_in, const int* in_sizes, int n_in,
                              void* d_out, int out_size, void* d_ws, size_t ws_size,
                              hipStream_t stream) {
    (void)n_in; (void)out_size; (void)d_ws; (void)ws_size;
    const float* t     = (const float*)d_in[0];
    const float* knots = (const float*)d_in[1];
    const float* a     = (const float*)d_in[2];
    const float* b     = (const float*)d_in[3];
    const float* c     = (const float*)d_in[4];
    const float* d     = (const float*)d_in[5];
    float* out = (float*)d_out;

    const int n_q     = in_sizes[0];
    const int n_rows  = in_sizes[2] / SPL_CH;   // 1023 segments
    const int max_idx = n_rows - 1;             // 1022 (reference's maxlen)

    const int grid = (n_q + SPL_TILE_Q - 1) / SPL_TILE_Q;
    NaturalCubicSpline_83966610637198_kernel<<<grid, SPL_BLOCK, 0, stream>>>(
        t, knots, a, b, c, d, out, n_q, max_idx);
}